// EfficientSelfAttention_82145544503391
// MI455X (gfx1250) — compile-verified
//
#include <hip/hip_runtime.h>
#include <hip/hip_bf16.h>

typedef float v2f __attribute__((ext_vector_type(2)));
typedef float v8f __attribute__((ext_vector_type(8)));

#define WMMA_F32(A, B, C) \
    __builtin_amdgcn_wmma_f32_16x16x4_f32(false, (A), false, (B), (short)0, (C), false, false)

#define LDSP 66  // padded row stride (floats), even -> keeps 8B alignment for b64 LDS loads

// ---------------------------------------------------------------------------
// Generic fp32 WMMA GEMM: out[M][N] = A[M][K] @ W[K][N] + bias[N]
// MODE 0: A is plain row-major with leading dim lda.
// MODE 1: A is the spatially-reduced gather of x (conv-as-GEMM), K=1024.
// Block = 128 threads (4 waves); block tile = 16 (M) x 64 (N); wave tile 16x16.
// ---------------------------------------------------------------------------
template <int MODE>
__global__ void gemm16_wmma(const float* __restrict__ A, int lda,
                            const float* __restrict__ W, int ldw,
                            const float* __restrict__ bias,
                            float* __restrict__ out, int ldo, int K) {
  __shared__ __align__(16) float a_tile[16][LDSP];   // [m][k]
  __shared__ __align__(16) float w_tile[64][LDSP];   // [n][k]  (W^T chunk)

  const int tid  = threadIdx.x;
  const int lane = tid & 31;
  const int wave = tid >> 5;
  const int m0     = blockIdx.x * 16;
  const int n0blk  = blockIdx.y * 64;
  const int nw     = wave * 16;

  const int mrow  = lane & 15;        // A-frag M / B-frag N index
  const int khalf = (lane >> 4) << 1; // K pair selected by lane half

  v8f acc = {0.f, 0.f, 0.f, 0.f, 0.f, 0.f, 0.f, 0.f};

  for (int kb = 0; kb < K; kb += 64) {
    __syncthreads();
    // ---- stage A tile (16 x 64) ----
    for (int e = tid; e < 16 * 64; e += 128) {
      int m = e >> 6, kk = e & 63;
      float val;
      if (MODE == 0) {
        val = A[(m0 + m) * lda + kb + kk];
      } else {
        // conv gather: row r -> (b, s); k -> (tap, ic)
        int r  = m0 + m;
        int b  = r >> 10, s = r & 1023;
        int ho = s >> 5,  wo = s & 31;
        int kg = kb + kk;
        int tap = kg >> 8, ic = kg & 255;
        int kh = tap >> 1, kw = tap & 1;
        int h2 = 2 * ho + kh, w2 = 2 * wo + kw;
        int cp = 4 * h2 + (w2 >> 4);          // source channel of x
        int np = ((w2 & 15) << 8) + ic;       // source position of x
        val = A[((b << 12) + np) * 256 + cp];
      }
      a_tile[m][kk] = val;
    }
    // ---- stage W^T tile (64 x 64): w_tile[n][k] = W[(kb+k)*ldw + n0blk+n] ----
    for (int e = tid; e < 64 * 64; e += 128) {
      int k = e >> 6, n = e & 63;
      w_tile[n][k] = W[(kb + k) * ldw + n0blk + n];
    }
    __syncthreads();

    #pragma unroll
    for (int ks = 0; ks < 16; ++ks) {
      v2f a = *(const v2f*)&a_tile[mrow][4 * ks + khalf];
      v2f b = *(const v2f*)&w_tile[nw + mrow][4 * ks + khalf];
      acc = WMMA_F32(a, b, acc);
    }
  }

  const int col     = n0blk + nw + mrow;
  const float bv    = bias[col];
  const int halfsel = (lane >> 4) << 3;  // D-matrix: vgpr v holds row v (lo half) / v+8 (hi half)
  #pragma unroll
  for (int v = 0; v < 8; ++v) {
    out[(m0 + v + halfsel) * ldo + col] = acc[v] + bv;
  }
}

// ---------------------------------------------------------------------------
// LayerNorm over 256-wide rows. One wave per row, 8 rows per 256-thread block.
// ---------------------------------------------------------------------------
__global__ void layernorm_kernel(const float* __restrict__ in,
                                 const float* __restrict__ sc,
                                 const float* __restrict__ bi,
                                 float* __restrict__ outp) {
  const int lane = threadIdx.x & 31;
  const int wave = threadIdx.x >> 5;
  const int row  = blockIdx.x * 8 + wave;
  const float* rp = in + row * 256;

  float4 a = *(const float4*)(rp + lane * 4);
  float4 b = *(const float4*)(rp + 128 + lane * 4);

  float s = a.x + a.y + a.z + a.w + b.x + b.y + b.z + b.w;
  #pragma unroll
  for (int m = 1; m < 32; m <<= 1) s += __shfl_xor(s, m, 32);
  const float mu = s * (1.0f / 256.0f);

  float d, vs = 0.f;
  d = a.x - mu; vs += d * d;  d = a.y - mu; vs += d * d;
  d = a.z - mu; vs += d * d;  d = a.w - mu; vs += d * d;
  d = b.x - mu; vs += d * d;  d = b.y - mu; vs += d * d;
  d = b.z - mu; vs += d * d;  d = b.w - mu; vs += d * d;
  #pragma unroll
  for (int m = 1; m < 32; m <<= 1) vs += __shfl_xor(vs, m, 32);
  const float inv = rsqrtf(vs * (1.0f / 256.0f) + 1e-6f);

  int c = lane * 4;
  float4 o;
  o.x = (a.x - mu) * inv * sc[c + 0] + bi[c + 0];
  o.y = (a.y - mu) * inv * sc[c + 1] + bi[c + 1];
  o.z = (a.z - mu) * inv * sc[c + 2] + bi[c + 2];
  o.w = (a.w - mu) * inv * sc[c + 3] + bi[c + 3];
  *(float4*)(outp + row * 256 + c) = o;
  c = 128 + lane * 4;
  o.x = (b.x - mu) * inv * sc[c + 0] + bi[c + 0];
  o.y = (b.y - mu) * inv * sc[c + 1] + bi[c + 1];
  o.z = (b.z - mu) * inv * sc[c + 2] + bi[c + 2];
  o.w = (b.w - mu) * inv * sc[c + 3] + bi[c + 3];
  *(float4*)(outp + row * 256 + c) = o;
}

// ---------------------------------------------------------------------------
// Flash attention, fp32 WMMA. Block = 128 threads (4 waves); each wave owns
// 16 query rows, block covers 64. KV length 1024 streamed in chunks of 64.
// qbuf:   [B*4096][256]  (q for head h in cols h*32..h*32+31)
// kvbuf:  [B*1024][512]  (k cols h*32+d, v cols 256+h*32+d)
// attnout:[B*4096][256]
// ---------------------------------------------------------------------------
__global__ void attn_kernel(const float* __restrict__ qbuf,
                            const float* __restrict__ kvbuf,
                            float* __restrict__ attnout) {
  __shared__ __align__(16) float k_tile[64][34];      // [kv][d]
  __shared__ __align__(16) float vt_tile[32][LDSP];   // [d][kv]
  __shared__ __align__(16) float p_tile[4][16][LDSP]; // per-wave P [qrow][kv]

  const int tid  = threadIdx.x;
  const int lane = tid & 31;
  const int wave = tid >> 5;
  const int id   = blockIdx.x;
  const int qblk = id & 63;
  const int bh   = id >> 6;
  const int h    = bh & 7;
  const int b    = bh >> 3;
  const int q0   = qblk * 64 + wave * 16;

  const int mrow  = lane & 15;
  const int khalf = (lane >> 4) << 1;
  const float scale = 0.17677669529663687f;  // 32^-0.5 folded into Q

  // Q A-fragments (d = 32 -> 8 k-steps), kept in registers for whole KV loop
  v2f qf[8];
  const float* qp = qbuf + (size_t)(b * 4096 + q0 + mrow) * 256 + h * 32;
  #pragma unroll
  for (int ks = 0; ks < 8; ++ks) {
    v2f t = *(const v2f*)(qp + 4 * ks + khalf);
    qf[ks] = t * scale;
  }

  v8f o0 = {0.f, 0.f, 0.f, 0.f, 0.f, 0.f, 0.f, 0.f};
  v8f o1 = {0.f, 0.f, 0.f, 0.f, 0.f, 0.f, 0.f, 0.f};
  float m_run[8], l_run[8];
  #pragma unroll
  for (int v = 0; v < 8; ++v) { m_run[v] = -__builtin_inff(); l_run[v] = 0.f; }

  const float* kvb = kvbuf + (size_t)b * 1024 * 512;

  for (int kv0 = 0; kv0 < 1024; kv0 += 64) {
    __syncthreads();
    for (int e = tid; e < 64 * 32; e += 128) {        // K rows as-is
      int kv = e >> 5, dd = e & 31;
      k_tile[kv][dd] = kvb[(kv0 + kv) * 512 + h * 32 + dd];
    }
    for (int e = tid; e < 64 * 32; e += 128) {        // V transposed
      int kv = e >> 5, dd = e & 31;
      vt_tile[dd][kv] = kvb[(kv0 + kv) * 512 + 256 + h * 32 + dd];
    }
    __syncthreads();

    // S = (Q*scale) K^T : four 16x16 tiles over kv
    v8f s[4];
    #pragma unroll
    for (int nt = 0; nt < 4; ++nt) {
      v8f acc = {0.f, 0.f, 0.f, 0.f, 0.f, 0.f, 0.f, 0.f};
      #pragma unroll
      for (int ks = 0; ks < 8; ++ks) {
        v2f bf = *(const v2f*)&k_tile[nt * 16 + mrow][4 * ks + khalf];
        acc = WMMA_F32(qf[ks], bf, acc);
      }
      s[nt] = acc;
    }

    // Online softmax; row of slot v is (v + 8*lane_half), stats replicated per half
    #pragma unroll
    for (int v = 0; v < 8; ++v) {
      float mx = fmaxf(fmaxf(s[0][v], s[1][v]), fmaxf(s[2][v], s[3][v]));
      #pragma unroll
      for (int msk = 1; msk < 16; msk <<= 1) mx = fmaxf(mx, __shfl_xor(mx, msk, 32));
      const float mnew = fmaxf(m_run[v], mx);
      const float corr = __expf(m_run[v] - mnew);
      float ls = 0.f;
      #pragma unroll
      for (int nt = 0; nt < 4; ++nt) {
        float p = __expf(s[nt][v] - mnew);
        s[nt][v] = p;
        ls += p;
      }
      #pragma unroll
      for (int msk = 1; msk < 16; msk <<= 1) ls += __shfl_xor(ls, msk, 32);
      l_run[v] = l_run[v] * corr + ls;
      m_run[v] = mnew;
      o0[v] *= corr;
      o1[v] *= corr;
    }

    // P -> per-wave LDS tile (D layout -> row-major), then reload as A-fragments
    const int rhalf = (lane >> 4) << 3;
    #pragma unroll
    for (int nt = 0; nt < 4; ++nt)
      #pragma unroll
      for (int v = 0; v < 8; ++v)
        p_tile[wave][v + rhalf][nt * 16 + mrow] = s[nt][v];
    asm volatile("s_wait_dscnt 0" ::: "memory");  // own-wave P writes visible

    // O += P V  (contraction over 64 kv -> 16 k-steps, two 16-wide d tiles)
    #pragma unroll
    for (int kp = 0; kp < 16; ++kp) {
      v2f af = *(const v2f*)&p_tile[wave][mrow][4 * kp + khalf];
      v2f b0 = *(const v2f*)&vt_tile[mrow][4 * kp + khalf];
      v2f b1 = *(const v2f*)&vt_tile[16 + mrow][4 * kp + khalf];
      o0 = WMMA_F32(af, b0, o0);
      o1 = WMMA_F32(af, b1, o1);
    }
  }

  // epilogue: divide by row sums, scatter to [B,N,C] with col = h*32 + d
  float* op = attnout + (size_t)(b * 4096 + q0) * 256 + h * 32;
  const int rhalf = (lane >> 4) << 3;
  #pragma unroll
  for (int v = 0; v < 8; ++v) {
    const float inv = 1.0f / l_run[v];
    op[(v + rhalf) * 256 + mrow]      = o0[v] * inv;
    op[(v + rhalf) * 256 + 16 + mrow] = o1[v] * inv;
  }
}

// ---------------------------------------------------------------------------
extern "C" void kernel_launch(void* const* d_in, const int* in_sizes, int n_in,
                              void* d_out, int out_size, void* d_ws, size_t ws_size,
                              hipStream_t stream) {
  (void)in_sizes; (void)n_in; (void)out_size; (void)ws_size;
  const float* x      = (const float*)d_in[0];
  const float* Wq     = (const float*)d_in[1];
  const float* bq     = (const float*)d_in[2];
  const float* Wconv  = (const float*)d_in[3];   // [2][2][256][256] == [1024][256]
  const float* bconv  = (const float*)d_in[4];
  const float* ln_s   = (const float*)d_in[5];
  const float* ln_b   = (const float*)d_in[6];
  const float* Wkv    = (const float*)d_in[7];   // [256][512]
  const float* bkv    = (const float*)d_in[8];
  const float* Wproj  = (const float*)d_in[9];
  const float* bproj  = (const float*)d_in[10];
  float* out = (float*)d_out;

  char* ws = (char*)d_ws;
  float* qbuf  = (float*)(ws);                       // 16 MB: [16384][256]
  float* kvbuf = (float*)(ws + (size_t)(16u << 20)); //  8 MB: [4096][512]
  float* attnb = (float*)(ws + (size_t)(24u << 20)); // 16 MB: [16384][256]
  float* convb = (float*)(ws + (size_t)(24u << 20)); //  4 MB (dead before attn)
  float* normb = (float*)(ws + (size_t)(28u << 20)); //  4 MB (dead before attn)

  // 1) spatial reduction conv (gathered GEMM, K=1024) + bias
  gemm16_wmma<1><<<dim3(256, 4), 128, 0, stream>>>(x, 0, Wconv, 256, bconv, convb, 256, 1024);
  // 2) LayerNorm
  layernorm_kernel<<<512, 256, 0, stream>>>(convb, ln_s, ln_b, normb);
  // 3) KV projection
  gemm16_wmma<0><<<dim3(256, 8), 128, 0, stream>>>(normb, 256, Wkv, 512, bkv, kvbuf, 512, 256);
  // 4) Q projection
  gemm16_wmma<0><<<dim3(1024, 4), 128, 0, stream>>>(x, 256, Wq, 256, bq, qbuf, 256, 256);
  // 5) fused flash attention
  attn_kernel<<<2048, 128, 0, stream>>>(qbuf, kvbuf, attnb);
  // 6) output projection
  gemm16_wmma<0><<<dim3(1024, 4), 128, 0, stream>>>(attnb, 256, Wproj, 256, bproj, out, /*ldo=*/256, /*K=*/256);
}